// CGMambaLayer_58909771432098
// MI455X (gfx1250) — compile-verified
//
#include <hip/hip_runtime.h>
#include <hip/hip_bf16.h>

typedef __attribute__((ext_vector_type(2))) float v2f;
typedef __attribute__((ext_vector_type(8))) float v8f;

// ---------------- problem constants (match reference) ----------------
static constexpr int BB   = 4;
static constexpr int HH   = 64;
static constexpr int WW   = 64;
static constexpr int CC   = 384;
static constexpr int GG   = 4;
static constexpr int DM   = 96;   // d_model per group
static constexpr int DI   = 96;   // d_inner (expand=1)
static constexpr int NK   = 4;    // scan directions
static constexpr int RR   = 6;    // dt_rank
static constexpr int RED  = 24;   // SE reduction
static constexpr int LL   = HH * WW;        // 4096
static constexpr int BL   = BB * LL;        // 16384
static constexpr int CHN  = 32;             // scan chunks
static constexpr int CLEN = LL / CHN;       // 128
static constexpr float EPSF = 1e-5f;

// ---------------- workspace layout (float units) ----------------
static constexpr size_t N_TOK = (size_t)BL * CC;       // 6,291,456
static constexpr size_t N_GBD = (size_t)GG * BL * DI;  // 6,291,456
static constexpr size_t N_DBL = (size_t)GG * BB * NK * LL * 8;
static constexpr size_t N_CHS = (size_t)GG * BB * NK * CHN * DI;

static constexpr size_t OFF_XN   = 0;                    // LN(x)
static constexpr size_t OFF_Z    = OFF_XN + N_TOK;       // gate half of in_proj
static constexpr size_t OFF_U    = OFF_Z + N_GBD;        // conv out; later gated y
static constexpr size_t OFF_XH   = OFF_U + N_GBD;        // in_proj x half; later y_merged; later xm
static constexpr size_t OFF_YALL = OFF_XH + N_GBD;       // out_proj concat (B,L,C)
static constexpr size_t OFF_DBL  = OFF_YALL + N_TOK;     // (G,B,K,L,8): dt(6),Bs,Cs
static constexpr size_t OFF_CHA  = OFF_DBL + N_DBL;      // chunk prod(dA)
static constexpr size_t OFF_CHH  = OFF_CHA + N_CHS;      // chunk h (from 0)
static constexpr size_t OFF_CHI  = OFF_CHH + N_CHS;      // chunk incoming h
static constexpr size_t OFF_ZCH  = OFF_CHI + N_CHS;      // (B,C) channel means
static constexpr size_t OFF_GATE = OFF_ZCH + (size_t)BB * CC;  // SE gate (B,C)
static constexpr size_t OFF_GIDX = OFF_GATE + (size_t)BB * CC; // int area (G,K,L)

__device__ __forceinline__ float sigm(float x) { return 1.0f / (1.0f + __expf(-x)); }
__device__ __forceinline__ float siluf(float x) { return x * sigm(x); }

// ---------------- K1: xn = LayerNorm(x) over C ----------------
__global__ void cgm_ln_x(const float* __restrict__ x, const float* __restrict__ nw,
                         const float* __restrict__ nb, float* __restrict__ xn) {
  int t = blockIdx.x;            // token (b*L+l)
  int tid = threadIdx.x;         // 128
  float v[3], s = 0.f, ss = 0.f;
#pragma unroll
  for (int i = 0; i < 3; ++i) {
    int c = tid + i * 128;
    float val = x[(size_t)t * CC + c];
    v[i] = val; s += val; ss += val * val;
  }
  __shared__ float red[128];
  red[tid] = s; __syncthreads();
  for (int off = 64; off > 0; off >>= 1) { if (tid < off) red[tid] += red[tid + off]; __syncthreads(); }
  float mean = red[0] / CC; __syncthreads();
  red[tid] = ss; __syncthreads();
  for (int off = 64; off > 0; off >>= 1) { if (tid < off) red[tid] += red[tid + off]; __syncthreads(); }
  float var = red[0] / CC - mean * mean;
  float inv = rsqrtf(var + EPSF);
#pragma unroll
  for (int i = 0; i < 3; ++i) {
    int c = tid + i * 128;
    xn[(size_t)t * CC + c] = (v[i] - mean) * inv * nw[c] + nb[c];
  }
}

// ---------------- K2: zch[b][c] = mean over tokens of xn ----------------
__global__ void cgm_zch(const float* __restrict__ xn, float* __restrict__ zch) {
  int idx = blockIdx.x * blockDim.x + threadIdx.x;
  if (idx >= BB * CC) return;
  int b = idx / CC, c = idx % CC;
  const float* p = xn + (size_t)b * LL * CC + c;
  float s = 0.f;
  for (int l = 0; l < LL; ++l) s += p[(size_t)l * CC];
  zch[idx] = s / (float)LL;
}

// ---------------- K3: SE gate ----------------
__global__ void cgm_se_gate(const float* __restrict__ zch, const float* __restrict__ f1w,
                            const float* __restrict__ f1b, const float* __restrict__ f2w,
                            const float* __restrict__ f2b, float* __restrict__ gate) {
  int b = blockIdx.x, t = threadIdx.x;   // 384 threads
  __shared__ float hid[RED];
  if (t < RED) {
    float s = f1b[t];
    for (int c = 0; c < CC; ++c) s += f1w[t * CC + c] * zch[b * CC + c];
    hid[t] = fmaxf(s, 0.f);
  }
  __syncthreads();
  float s = f2b[t];
#pragma unroll
  for (int r = 0; r < RED; ++r) s += f2w[t * RED + r] * hid[r];
  gate[b * CC + t] = sigm(s);
}

// ---------------- K4: generate scan permutations (spirals) ----------------
__device__ void gen_spiral(int n0, int n1, bool cw, bool transposed, bool rev, int* __restrict__ out) {
  const int Ltot = n0 * n1;
  int i = 0;
  int top = 0, bottom = n0 - 1, left = 0, right = n1 - 1;
  auto emit = [&](int r, int c) {
    int v0 = r * n1 + c;
    int v = transposed ? (v0 % HH) * WW + (v0 / HH) : v0;
    int j = rev ? (Ltot - 1 - i) : i;
    out[j] = v;
    ++i;
  };
  while (top <= bottom && left <= right) {
    if (cw) {
      for (int c = left; c <= right; ++c) emit(top, c);
      for (int r = top + 1; r <= bottom; ++r) emit(r, right);
      if (bottom > top) for (int c = right - 1; c >= left; --c) emit(bottom, c);
      if (right > left) for (int r = bottom - 1; r > top; --r) emit(r, left);
    } else {
      for (int r = top; r <= bottom; ++r) emit(r, left);
      for (int c = left + 1; c <= right; ++c) emit(bottom, c);
      if (right > left) for (int r = bottom - 1; r >= top; --r) emit(r, right);
      if (bottom > top) for (int c = right - 1; c > left; --c) emit(top, c);
    }
    ++top; --bottom; ++left; --right;
  }
}

__global__ void cgm_gen_perms(int* __restrict__ gidx) {
  int t = threadIdx.x;
  if (t >= GG * NK) return;
  int g = t >> 2, k = t & 3;
  bool cw = (g < 2);
  bool conv = (g & 1);
  bool transposed = (k & 1);
  bool rev = conv ^ (k >= 2);
  int* out = gidx + (size_t)(g * NK + k) * LL;
  if (!transposed) gen_spiral(HH, WW, cw, false, rev, out);
  else             gen_spiral(WW, HH, cw, true,  rev, out);
}

// ---------------- WMMA f32 16x16x4 helpers ----------------
// A 16x4: lane r=lane&15 holds row r; half=lane>>4 selects K pair {0,1}/{2,3}.
// B 4x16: lane n=lane&15 holds col n; same K-pair split.
// C/D 16x16 v8f: element v -> row (half*8+v), col (lane&15).

// ---------------- K5: in_proj GEMM (per group, 16384x192x96) ----------------
__global__ void cgm_in_proj(const float* __restrict__ xn, const float* __restrict__ ipw,
                            float* __restrict__ xh, float* __restrict__ zb) {
  int lane = threadIdx.x & 31, wave = threadIdx.x >> 5;
  int r = lane & 15, half = lane >> 4;
  int m0 = (blockIdx.x * 4 + wave) * 16;
  int n0 = blockIdx.y * 16;
  int g = blockIdx.z;
  const float* arow = xn + (size_t)(m0 + r) * CC + g * DM;
  const float* wrow = ipw + ((size_t)g * 2 * DI + (n0 + r)) * DM;
  v8f acc = {};
  for (int kb = 0; kb < DM; kb += 4) {
    int ko = kb + half * 2;
    v2f a;  a.x = arow[ko];  a.y = arow[ko + 1];
    v2f bv; bv.x = wrow[ko]; bv.y = wrow[ko + 1];
    acc = __builtin_amdgcn_wmma_f32_16x16x4_f32(false, a, false, bv, (short)0, acc, false, false);
  }
#pragma unroll
  for (int v = 0; v < 8; ++v) {
    int m = m0 + (half << 3) + v;
    int e = n0 + r;
    size_t base = ((size_t)g * BL + m) * DI;
    float val = acc[v];
    if (e < DI) xh[base + e] = val;
    else        zb[base + e - DI] = val;
  }
}

// ---------------- K6: depthwise 3x3 conv + bias + silu ----------------
__global__ void cgm_dwconv(const float* __restrict__ xh, const float* __restrict__ cw,
                           const float* __restrict__ cb, float* __restrict__ u) {
  size_t idx = (size_t)blockIdx.x * blockDim.x + threadIdx.x;
  if (idx >= (size_t)GG * BL * DI) return;
  int d = (int)(idx % DI);
  size_t gm = idx / DI;
  int g = (int)(gm / BL);
  int m = (int)(gm % BL);
  int b = m / LL, l = m % LL;
  int y = l / WW, x = l % WW;
  float s = cb[g * DI + d];
#pragma unroll
  for (int ky = 0; ky < 3; ++ky) {
    int yy = y + ky - 1;
    if (yy < 0 || yy >= HH) continue;
#pragma unroll
    for (int kx = 0; kx < 3; ++kx) {
      int xx = x + kx - 1;
      if (xx < 0 || xx >= WW) continue;
      float wv = cw[((g * DI + d) * 3 + ky) * 3 + kx];
      s += wv * xh[((size_t)g * BL + (size_t)b * LL + yy * WW + xx) * DI + d];
    }
  }
  u[idx] = siluf(s);
}

// ---------------- K7: x_proj -> dbl (dt[6], Bs, Cs) per scan position ----------------
__global__ void cgm_xproj(const float* __restrict__ u, const int* __restrict__ gidx,
                          const float* __restrict__ xpw, float* __restrict__ dbl) {
  int idx = blockIdx.x * blockDim.x + threadIdx.x;   // G*B*NK*L = 262144
  if (idx >= GG * BB * NK * LL) return;
  int l = idx % LL;
  int k = (idx / LL) % NK;
  int b = (idx / (LL * NK)) % BB;
  int g = idx / (LL * NK * BB);
  int pos = gidx[(size_t)(g * NK + k) * LL + l];
  const float* ur = u + ((size_t)g * BL + (size_t)b * LL + pos) * DI;
  const float* wp = xpw + (size_t)(g * NK + k) * 8 * DI;
  float acc[8] = {0.f, 0.f, 0.f, 0.f, 0.f, 0.f, 0.f, 0.f};
  for (int d = 0; d < DI; ++d) {
    float xv = ur[d];
#pragma unroll
    for (int c = 0; c < 8; ++c) acc[c] += wp[c * DI + d] * xv;
  }
  float* o = dbl + (size_t)idx * 8;
#pragma unroll
  for (int c = 0; c < 8; ++c) o[c] = acc[c];
}

// ---------------- scan step core ----------------
__device__ __forceinline__ void scan_params(const float* dtw, const float* dtb,
                                            const float* alog, const float* dss,
                                            int gk, int d, float* wr, float& bias,
                                            float& Av, float& Dv) {
#pragma unroll
  for (int r = 0; r < RR; ++r) wr[r] = dtw[((size_t)gk * DI + d) * RR + r];
  bias = dtb[gk * DI + d];
  Av = -__expf(alog[gk * DI + d]);   // DSTATE == 1
  Dv = dss[gk * DI + d];
}

// K8a: per-chunk scan from zero state -> (prod dA, h_end)
__global__ void cgm_scan_pass1(const float* __restrict__ u, const float* __restrict__ dbl,
                               const int* __restrict__ gidx, const float* __restrict__ dtw,
                               const float* __restrict__ dtb, const float* __restrict__ alog,
                               const float* __restrict__ dss, float* __restrict__ cha,
                               float* __restrict__ chh) {
  int tid = blockIdx.x * blockDim.x + threadIdx.x;   // G*B*NK*CHN*DI = 196608
  if (tid >= GG * BB * NK * CHN * DI) return;
  int d = tid % DI;
  int ch = (tid / DI) % CHN;
  int gbk = tid / (DI * CHN);           // (g*B+b)*NK + k
  int k = gbk % NK;
  int b = (gbk / NK) % BB;
  int g = gbk / (NK * BB);
  int gk = g * NK + k;
  float wr[RR], bias, Av, Dv;
  scan_params(dtw, dtb, alog, dss, gk, d, wr, bias, Av, Dv);
  const int* gi = gidx + (size_t)gk * LL;
  const float* ub = u + ((size_t)g * BL + (size_t)b * LL) * DI + d;
  const float* db = dbl + (size_t)gbk * LL * 8;
  float h = 0.f, aP = 1.f;
  int l0 = ch * CLEN;
  for (int l = l0; l < l0 + CLEN; ++l) {
    int pos = gi[l];
    float xs = ub[(size_t)pos * DI];
    const float* dr = db + (size_t)l * 8;
    float dt = bias;
#pragma unroll
    for (int r = 0; r < RR; ++r) dt += wr[r] * dr[r];
    float delta = (dt > 20.f) ? dt : __logf(1.f + __expf(dt));
    float dA = __expf(delta * Av);
    h = dA * h + delta * xs * dr[6];
    aP *= dA;
  }
  cha[tid] = aP;
  chh[tid] = h;
}

// K8b: prefix over chunks -> incoming state per chunk
__global__ void cgm_scan_pass2(const float* __restrict__ cha, const float* __restrict__ chh,
                               float* __restrict__ chi) {
  int tid = blockIdx.x * blockDim.x + threadIdx.x;   // G*B*NK*DI = 6144
  if (tid >= GG * BB * NK * DI) return;
  int d = tid % DI;
  int gbk = tid / DI;
  float h = 0.f;
  for (int ch = 0; ch < CHN; ++ch) {
    size_t idx = ((size_t)gbk * CHN + ch) * DI + d;
    chi[idx] = h;
    h = cha[idx] * h + chh[idx];
  }
}

// K8c: replay chunks from incoming state, emit y, scatter (cross-merge) via gidx
__global__ void cgm_scan_pass3(const float* __restrict__ u, const float* __restrict__ dbl,
                               const int* __restrict__ gidx, const float* __restrict__ dtw,
                               const float* __restrict__ dtb, const float* __restrict__ alog,
                               const float* __restrict__ dss, const float* __restrict__ chi,
                               float* __restrict__ ym) {
  int tid = blockIdx.x * blockDim.x + threadIdx.x;
  if (tid >= GG * BB * NK * CHN * DI) return;
  int d = tid % DI;
  int ch = (tid / DI) % CHN;
  int gbk = tid / (DI * CHN);
  int k = gbk % NK;
  int b = (gbk / NK) % BB;
  int g = gbk / (NK * BB);
  int gk = g * NK + k;
  float wr[RR], bias, Av, Dv;
  scan_params(dtw, dtb, alog, dss, gk, d, wr, bias, Av, Dv);
  const int* gi = gidx + (size_t)gk * LL;
  const float* ub = u + ((size_t)g * BL + (size_t)b * LL) * DI + d;
  float* yb = ym + ((size_t)g * BL + (size_t)b * LL) * DI + d;
  const float* db = dbl + (size_t)gbk * LL * 8;
  float h = chi[tid];
  int l0 = ch * CLEN;
  for (int l = l0; l < l0 + CLEN; ++l) {
    int pos = gi[l];
    float xs = ub[(size_t)pos * DI];
    const float* dr = db + (size_t)l * 8;
    float dt = bias;
#pragma unroll
    for (int r = 0; r < RR; ++r) dt += wr[r] * dr[r];
    float delta = (dt > 20.f) ? dt : __logf(1.f + __expf(dt));
    float dA = __expf(delta * Av);
    h = dA * h + delta * xs * dr[6];
    float yv = h * dr[7] + Dv * xs;
    atomicAdd(&yb[(size_t)pos * DI], yv);
  }
}

// ---------------- K9: out_norm LN over DI + silu(z) gate ----------------
__global__ void cgm_outnorm_gate(const float* __restrict__ ym, const float* __restrict__ zb,
                                 const float* __restrict__ onw, const float* __restrict__ onb,
                                 float* __restrict__ gated) {
  int wave = threadIdx.x >> 5, lane = threadIdx.x & 31;
  size_t row = (size_t)blockIdx.x * 8 + wave;     // < G*BL
  int g = (int)(row / BL);
  const float* yr = ym + row * DI;
  float v[3];
  float s = 0.f, ss = 0.f;
#pragma unroll
  for (int i = 0; i < 3; ++i) {
    float val = yr[lane + i * 32];
    v[i] = val; s += val; ss += val * val;
  }
  for (int off = 16; off > 0; off >>= 1) { s += __shfl_down(s, off); ss += __shfl_down(ss, off); }
  s = __shfl(s, 0); ss = __shfl(ss, 0);
  float mean = s / DI;
  float var = ss / DI - mean * mean;
  float inv = rsqrtf(var + EPSF);
  const float* zr = zb + row * DI;
  float* gr = gated + row * DI;
#pragma unroll
  for (int i = 0; i < 3; ++i) {
    int d = lane + i * 32;
    float zv = zr[d];
    gr[d] = ((v[i] - mean) * inv * onw[g * DI + d] + onb[g * DI + d]) * siluf(zv);
  }
}

// ---------------- K10: out_proj GEMM (per group, 16384x96x96) ----------------
__global__ void cgm_out_proj(const float* __restrict__ gated, const float* __restrict__ opw,
                             float* __restrict__ yall) {
  int lane = threadIdx.x & 31, wave = threadIdx.x >> 5;
  int r = lane & 15, half = lane >> 4;
  int m0 = (blockIdx.x * 4 + wave) * 16;
  int n0 = blockIdx.y * 16;
  int g = blockIdx.z;
  const float* arow = gated + ((size_t)g * BL + m0 + r) * DI;
  const float* wrow = opw + ((size_t)g * DM + (n0 + r)) * DI;
  v8f acc = {};
  for (int kb = 0; kb < DI; kb += 4) {
    int ko = kb + half * 2;
    v2f a;  a.x = arow[ko];  a.y = arow[ko + 1];
    v2f bv; bv.x = wrow[ko]; bv.y = wrow[ko + 1];
    acc = __builtin_amdgcn_wmma_f32_16x16x4_f32(false, a, false, bv, (short)0, acc, false, false);
  }
#pragma unroll
  for (int v = 0; v < 8; ++v) {
    int m = m0 + (half << 3) + v;
    int n = n0 + r;
    yall[(size_t)m * CC + g * DM + n] = acc[v];
  }
}

// ---------------- K11: skip*SE-gate elementwise + shared LN ----------------
__global__ void cgm_premix_ln(const float* __restrict__ yall, const float* __restrict__ xn,
                              const float* __restrict__ gate, const float* __restrict__ skipp,
                              const float* __restrict__ nw, const float* __restrict__ nb,
                              float* __restrict__ xm) {
  int t = blockIdx.x;
  int tid = threadIdx.x;   // 128
  int b = t / LL;
  float skip = skipp[0];
  float v[3], s = 0.f, ss = 0.f;
#pragma unroll
  for (int i = 0; i < 3; ++i) {
    int c = tid + i * 128;
    float val = yall[(size_t)t * CC + c] * skip * xn[(size_t)t * CC + c] * gate[b * CC + c];
    v[i] = val; s += val; ss += val * val;
  }
  __shared__ float red[128];
  red[tid] = s; __syncthreads();
  for (int off = 64; off > 0; off >>= 1) { if (tid < off) red[tid] += red[tid + off]; __syncthreads(); }
  float mean = red[0] / CC; __syncthreads();
  red[tid] = ss; __syncthreads();
  for (int off = 64; off > 0; off >>= 1) { if (tid < off) red[tid] += red[tid + off]; __syncthreads(); }
  float var = red[0] / CC - mean * mean;
  float inv = rsqrtf(var + EPSF);
#pragma unroll
  for (int i = 0; i < 3; ++i) {
    int c = tid + i * 128;
    xm[(size_t)t * CC + c] = (v[i] - mean) * inv * nw[c] + nb[c];
  }
}

// ---------------- K12: final proj GEMM + alpha*(..)+identity ----------------
__global__ void cgm_final_proj(const float* __restrict__ xm, const float* __restrict__ pw,
                               const float* __restrict__ pb, const float* __restrict__ alphap,
                               const float* __restrict__ x, float* __restrict__ out) {
  int lane = threadIdx.x & 31, wave = threadIdx.x >> 5;
  int r = lane & 15, half = lane >> 4;
  int m0 = (blockIdx.x * 4 + wave) * 16;
  int n0 = blockIdx.y * 16;
  const float* arow = xm + (size_t)(m0 + r) * CC;
  const float* wrow = pw + (size_t)(n0 + r) * CC;
  v8f acc = {};
  for (int kb = 0; kb < CC; kb += 4) {
    int ko = kb + half * 2;
    v2f a;  a.x = arow[ko];  a.y = arow[ko + 1];
    v2f bv; bv.x = wrow[ko]; bv.y = wrow[ko + 1];
    acc = __builtin_amdgcn_wmma_f32_16x16x4_f32(false, a, false, bv, (short)0, acc, false, false);
  }
  float al = alphap[0];
#pragma unroll
  for (int v = 0; v < 8; ++v) {
    int m = m0 + (half << 3) + v;
    int n = n0 + r;
    size_t i = (size_t)m * CC + n;
    out[i] = al * (acc[v] + pb[n]) + x[i];
  }
}

// ---------------- host launcher ----------------
extern "C" void kernel_launch(void* const* d_in, const int* in_sizes, int n_in,
                              void* d_out, int out_size, void* d_ws, size_t ws_size,
                              hipStream_t stream) {
  (void)in_sizes; (void)n_in; (void)out_size; (void)ws_size;
  const float* x      = (const float*)d_in[0];
  // d_in[1]=H, d_in[2]=W: fixed at 64 in the reference; hardcoded.
  const float* norm_w = (const float*)d_in[3];
  const float* norm_b = (const float*)d_in[4];
  const float* fc1_w  = (const float*)d_in[5];
  const float* fc1_b  = (const float*)d_in[6];
  const float* fc2_w  = (const float*)d_in[7];
  const float* fc2_b  = (const float*)d_in[8];
  const float* ipw    = (const float*)d_in[9];
  const float* conv_w = (const float*)d_in[10];
  const float* conv_b = (const float*)d_in[11];
  const float* xpw    = (const float*)d_in[12];
  const float* dtw    = (const float*)d_in[13];
  const float* dtb    = (const float*)d_in[14];
  const float* alogs  = (const float*)d_in[15];
  const float* dss    = (const float*)d_in[16];
  const float* onw    = (const float*)d_in[17];
  const float* onb    = (const float*)d_in[18];
  const float* opw    = (const float*)d_in[19];
  const float* pw     = (const float*)d_in[20];
  const float* pb     = (const float*)d_in[21];
  const float* skip   = (const float*)d_in[22];
  const float* alpha  = (const float*)d_in[23];
  float* out = (float*)d_out;

  float* ws = (float*)d_ws;
  float* xn    = ws + OFF_XN;
  float* zbuf  = ws + OFF_Z;
  float* ubuf  = ws + OFF_U;     // conv out; later gated
  float* xhbuf = ws + OFF_XH;    // in_proj x-half; later y_merged; later xm
  float* yall  = ws + OFF_YALL;
  float* dbl   = ws + OFF_DBL;
  float* cha   = ws + OFF_CHA;
  float* chh   = ws + OFF_CHH;
  float* chi   = ws + OFF_CHI;
  float* zch   = ws + OFF_ZCH;
  float* gate  = ws + OFF_GATE;
  int*   gidx  = (int*)(ws + OFF_GIDX);

  // 1) shared LN
  cgm_ln_x<<<BL, 128, 0, stream>>>(x, norm_w, norm_b, xn);
  // 2-3) SE branch
  cgm_zch<<<(BB * CC + 255) / 256, 256, 0, stream>>>(xn, zch);
  cgm_se_gate<<<BB, CC, 0, stream>>>(zch, fc1_w, fc1_b, fc2_w, fc2_b, gate);
  // 4) scan permutations
  cgm_gen_perms<<<1, 32, 0, stream>>>(gidx);
  // 5) in_proj (WMMA)
  cgm_in_proj<<<dim3(BL / 64, (2 * DI) / 16, GG), 128, 0, stream>>>(xn, ipw, xhbuf, zbuf);
  // 6) depthwise conv + silu
  cgm_dwconv<<<(unsigned)((size_t)GG * BL * DI / 256), 256, 0, stream>>>(xhbuf, conv_w, conv_b, ubuf);
  // y_merged accumulator (reuses xh buffer) must start at zero
  hipMemsetAsync(xhbuf, 0, N_GBD * sizeof(float), stream);
  // 7) x_proj small projections
  cgm_xproj<<<(GG * BB * NK * LL) / 256, 256, 0, stream>>>(ubuf, gidx, xpw, dbl);
  // 8) chunked selective scan (3 passes)
  cgm_scan_pass1<<<(GG * BB * NK * CHN * DI) / 256, 256, 0, stream>>>(ubuf, dbl, gidx, dtw, dtb,
                                                                      alogs, dss, cha, chh);
  cgm_scan_pass2<<<(GG * BB * NK * DI) / 256, 256, 0, stream>>>(cha, chh, chi);
  cgm_scan_pass3<<<(GG * BB * NK * CHN * DI) / 256, 256, 0, stream>>>(ubuf, dbl, gidx, dtw, dtb,
                                                                      alogs, dss, chi, xhbuf);
  // 9) out_norm + silu(z) gate (writes into u buffer, now free)
  cgm_outnorm_gate<<<(GG * BL) / 8, 256, 0, stream>>>(xhbuf, zbuf, onw, onb, ubuf);
  // 10) out_proj (WMMA) -> concat groups
  cgm_out_proj<<<dim3(BL / 64, DM / 16, GG), 128, 0, stream>>>(ubuf, opw, yall);
  // 11) skip * ximg * SE gate + shared LN (writes into xh buffer, now free)
  cgm_premix_ln<<<BL, 128, 0, stream>>>(yall, xn, gate, skip, norm_w, norm_b, xhbuf);
  // 12) final projection (WMMA) + alpha*() + identity
  cgm_final_proj<<<dim3(BL / 64, CC / 16), 128, 0, stream>>>(xhbuf, pw, pb, alpha, x, out);
}